// SubtokenToToken_51007031607984
// MI455X (gfx1250) — compile-verified
//
#include <hip/hip_runtime.h>
#include <stdint.h>

// Problem shape fixed by the reference's setup_inputs().
#define BATCH 32
#define SEQ   4096
#define FEAT  256

#define ROWS_PER_WAVE  4
#define WAVES_PER_BLK  8
#define ROWS_PER_BLK   (ROWS_PER_WAVE * WAVES_PER_BLK)   // 32 rows, 32KB LDS

typedef float v4f __attribute__((ext_vector_type(4)));   // native vector for NT stores

// ---------------------------------------------------------------------------
// Kernel 1: per-batch-row exclusive scan of the mask.
// Builds the inverse map: src_idx[b*T + t] = s  for the t-th masked position s
// in row b, and count[b] = number of masked positions in row b.
// One 256-thread block per batch row; 16 mask elements per thread.
// ---------------------------------------------------------------------------
__global__ void mask_scan_kernel(const int* __restrict__ mask,
                                 int* __restrict__ src_idx,
                                 int* __restrict__ count,
                                 int T) {
  const int b   = blockIdx.x;
  const int tid = threadIdx.x;            // 0..255
  constexpr int PER = SEQ / 256;          // 16
  const int s0 = tid * PER;
  const int* m = mask + (size_t)b * SEQ;

  int local[PER];
  int sum = 0;
#pragma unroll
  for (int i = 0; i < PER; ++i) { local[i] = m[s0 + i]; sum += local[i]; }

  __shared__ int sc[256];
  sc[tid] = sum;
  __syncthreads();
  // Hillis-Steele inclusive scan over the 256 per-thread totals.
  for (int off = 1; off < 256; off <<= 1) {
    int v   = sc[tid];
    int add = (tid >= off) ? sc[tid - off] : 0;
    __syncthreads();
    sc[tid] = v + add;
    __syncthreads();
  }
  int pos = (tid == 0) ? 0 : sc[tid - 1];   // exclusive prefix for this thread
  if (tid == 255) count[b] = sc[255];

#pragma unroll
  for (int i = 0; i < PER; ++i) {
    if (local[i]) {
      if (pos < T) src_idx[(size_t)b * T + pos] = s0 + i;
      ++pos;
    }
  }
}

// ---------------------------------------------------------------------------
// Kernel 2: output-row pass, pipelined through the gfx1250 async global<->LDS
// DMA pipe. Each wave32 owns 4 complete 1KB output rows (32B per lane).
//   Phase 1: issue up to 8 async b128 loads back-to-back (ASYNCcnt tracks
//            them; up to 4KB in flight per wave) -> one s_wait_asynccnt 0.
//   Phase 2: async b128 stores from LDS for data rows; NT zero stores for
//            pad rows. s_endpgm's implicit wait-idle drains the stores.
// Every output byte written exactly once; only masked input rows read.
// Non-temporal hints: pure streaming traffic, no L2 reuse to protect.
// ---------------------------------------------------------------------------
__global__ void pack_rows_kernel(const float* __restrict__ units,
                                 const int*  __restrict__ src_idx,
                                 const int*  __restrict__ count,
                                 float* __restrict__ out,
                                 int T, int nrows) {
  __shared__ float tile[ROWS_PER_BLK][FEAT];     // 32 rows x 1KB = 32KB
  const int tid   = threadIdx.x;                 // 0..255
  const int w     = tid >> 5;                    // wave id 0..7
  const int lane  = tid & 31;                    // 0..31, 32 bytes per lane
  const int rbase = blockIdx.x * ROWS_PER_BLK + w * ROWS_PER_WAVE;

  int is_data[ROWS_PER_WAVE];

  // ---- Phase 1: batch async loads (global -> LDS) ----
#pragma unroll
  for (int i = 0; i < ROWS_PER_WAVE; ++i) {
    const int r = rbase + i;                     // flat output row = b*T + t
    is_data[i] = 0;
    if (r < nrows) {                             // wave-uniform
      const int b = r / T;
      const int t = r - b * T;
      if (t < count[b]) {                        // wave-uniform
        is_data[i] = 1;
        const int s = src_idx[(size_t)b * T + t];
        const float* srcp = units + ((size_t)b * SEQ + (size_t)s) * FEAT;
        uint64_t ga = (uint64_t)(uintptr_t)srcp + (uint64_t)(lane * 32);
        uint32_t la = (uint32_t)(uintptr_t)&tile[w * ROWS_PER_WAVE + i][lane * 8];
        asm volatile("global_load_async_to_lds_b128 %0, %1, off th:TH_LOAD_NT"
                     :: "v"(la), "v"(ga) : "memory");
        asm volatile("global_load_async_to_lds_b128 %0, %1, off offset:16 th:TH_LOAD_NT"
                     :: "v"(la), "v"(ga) : "memory");
      }
    }
  }

  // Single wait for all of this wave's outstanding async loads.
  asm volatile("s_wait_asynccnt 0" ::: "memory");

  // ---- Phase 2: async stores (LDS -> global) / zero padding ----
#pragma unroll
  for (int i = 0; i < ROWS_PER_WAVE; ++i) {
    const int r = rbase + i;
    if (r >= nrows) continue;
    float* dst = out + (size_t)r * FEAT;
    if (is_data[i]) {
      uint32_t la = (uint32_t)(uintptr_t)&tile[w * ROWS_PER_WAVE + i][lane * 8];
      uint64_t oa = (uint64_t)(uintptr_t)dst + (uint64_t)(lane * 32);
      asm volatile("global_store_async_from_lds_b128 %0, %1, off th:TH_STORE_NT"
                   :: "v"(oa), "v"(la) : "memory");
      asm volatile("global_store_async_from_lds_b128 %0, %1, off offset:16 th:TH_STORE_NT"
                   :: "v"(oa), "v"(la) : "memory");
    } else {
      v4f z = (v4f)(0.0f);
      __builtin_nontemporal_store(z, (v4f*)(dst + lane * 8));
      __builtin_nontemporal_store(z, (v4f*)(dst + lane * 8 + 4));
    }
  }
}

// ---------------------------------------------------------------------------
// Host-side launcher. T is recovered from out_size (out = [B, T, F]).
// Workspace layout: count[64] (padded) then src_idx[B*T]  (<= 512KB + 256B).
// ---------------------------------------------------------------------------
extern "C" void kernel_launch(void* const* d_in, const int* in_sizes, int n_in,
                              void* d_out, int out_size, void* d_ws, size_t ws_size,
                              hipStream_t stream) {
  const float* units = (const float*)d_in[0];
  const int*   mask  = (const int*)d_in[1];
  float*       out   = (float*)d_out;

  const int T = out_size / (BATCH * FEAT);
  if (T <= 0) return;

  int* count   = (int*)d_ws;
  int* src_idx = count + 64;  // keep src_idx 256B-aligned past the counts

  mask_scan_kernel<<<BATCH, 256, 0, stream>>>(mask, src_idx, count, T);

  const int nrows = BATCH * T;
  pack_rows_kernel<<<(nrows + ROWS_PER_BLK - 1) / ROWS_PER_BLK, 256, 0, stream>>>(
      units, src_idx, count, out, T, nrows);
}